// NLM2D_13116830122307
// MI455X (gfx1250) — compile-verified
//
#include <hip/hip_runtime.h>

// Fast-mode non-local means, 8x4x768x768 fp32.
// Tile in LDS (loaded once via CDNA5 async global->LDS gather with reflect
// padding folded into per-lane addresses); 169 shifts computed entirely from
// LDS. Per shift: fused squared-diff + horizontal 5-sum stage (per-thread
// sliding window, power-of-two index math), ping-pong row-sum buffer so only
// ONE barrier per shift is required, then a per-thread vertical sliding 5-sum,
// v_exp_f32 weight, and FMA accumulation in registers.

#define IMG       768
#define TILE_W    32
#define TILE_H    32
#define HALO      8                 // patch radius (2) + search radius (6)
#define PAD_W     (TILE_W + 2*HALO) // 48
#define PAD_H     (TILE_H + 2*HALO) // 48
#define TSTRIDE   49                // padded LDS stride (floats)
#define NTHREADS  256

__global__ __launch_bounds__(NTHREADS)
void nlm2d_kernel(const float* __restrict__ xin, float* __restrict__ out) {
    __shared__ float tile[PAD_H][TSTRIDE];  // reflected input tile (9.4 KB)
    __shared__ float rs[2][36][32];         // ping-pong horizontal 5-sums (9.2 KB)

    const int tid = threadIdx.x;
    const int x0  = blockIdx.x * TILE_W;
    const int y0  = blockIdx.y * TILE_H;
    const int img = blockIdx.z;

    const float* src = xin + (size_t)img * (IMG * IMG);

    // ---- async gather of padded tile into LDS (reflect at image borders) ----
    unsigned ldsBase = (unsigned)(size_t)&tile[0][0];
    for (int i = tid; i < PAD_H * PAD_W; i += NTHREADS) {  // 2304 = 9*256 exactly
        int r = i / PAD_W;
        int c = i - r * PAD_W;
        int gy = y0 - HALO + r;
        gy = (gy < 0) ? -gy : ((gy >= IMG) ? (2 * IMG - 2 - gy) : gy);
        int gx = x0 - HALO + c;
        gx = (gx < 0) ? -gx : ((gx >= IMG) ? (2 * IMG - 2 - gx) : gx);
        unsigned long long ga = (unsigned long long)(size_t)(src + gy * IMG + gx);
        unsigned lo = ldsBase + (unsigned)((r * TSTRIDE + c) * 4);
        asm volatile("global_load_async_to_lds_b32 %0, %1, off"
                     :: "v"(lo), "v"(ga) : "memory");
    }
    asm volatile("s_wait_asynccnt 0x0" ::: "memory");
    __syncthreads();

    const int lx = tid & 31;        // output column within tile
    const int ys = (tid >> 5) << 2; // first of 4 output rows owned by this thread

    float num0 = 0.f, num1 = 0.f, num2 = 0.f, num3 = 0.f;
    float den0 = 0.f, den1 = 0.f, den2 = 0.f, den3 = 0.f;

    const float kExp = -23.083120654223414f; // -16 * log2(e); w = exp(-16*sum25)
    int par = 0;

    #pragma unroll 1
    for (int dy = -6; dy <= 6; ++dy) {
        #pragma unroll 1
        for (int dx = -6; dx <= 6; ++dx) {
            // ---- S12: fused squared-diff + horizontal sliding 5-sum ----
            // 36 rows x 8 groups of 4 consecutive output columns = 288 units.
            for (int i = tid; i < 36 * 8; i += NTHREADS) {
                int r  = i >> 3;          // rs row (output y = r-2)
                int c0 = (i & 7) << 2;    // first of 4 output columns
                const float* bp = &tile[r + 6][c0 + 6];
                const float* sp = &tile[r + 6 + dy][c0 + 6 + dx];
                float t;
                t = bp[0] - sp[0]; float q0 = t * t;
                t = bp[1] - sp[1]; float q1 = t * t;
                t = bp[2] - sp[2]; float q2 = t * t;
                t = bp[3] - sp[3]; float q3 = t * t;
                t = bp[4] - sp[4]; float q4 = t * t;
                t = bp[5] - sp[5]; float q5 = t * t;
                t = bp[6] - sp[6]; float q6 = t * t;
                t = bp[7] - sp[7]; float q7 = t * t;
                float s = q0 + q1 + q2 + q3 + q4;
                rs[par][r][c0 + 0] = s;
                s += q5 - q0;
                rs[par][r][c0 + 1] = s;
                s += q6 - q1;
                rs[par][r][c0 + 2] = s;
                s += q7 - q2;
                rs[par][r][c0 + 3] = s;
            }
            __syncthreads();   // the ONLY barrier per shift (ping-pong rs)

            // ---- S3: vertical sliding 5-sum, weight, accumulate (4 rows) ----
            float r0 = rs[par][ys + 0][lx], r1 = rs[par][ys + 1][lx];
            float r2 = rs[par][ys + 2][lx], r3 = rs[par][ys + 3][lx];
            float r4 = rs[par][ys + 4][lx], r5 = rs[par][ys + 5][lx];
            float r6 = rs[par][ys + 6][lx], r7 = rs[par][ys + 7][lx];
            float s0 = r0 + r1 + r2 + r3 + r4;
            float s1 = s0 + r5 - r0;
            float s2 = s1 + r6 - r1;
            float s3 = s2 + r7 - r2;
            float w0 = __builtin_amdgcn_exp2f(s0 * kExp);
            float w1 = __builtin_amdgcn_exp2f(s1 * kExp);
            float w2 = __builtin_amdgcn_exp2f(s2 * kExp);
            float w3 = __builtin_amdgcn_exp2f(s3 * kExp);
            int vr = ys + dy + HALO;
            int vc = lx + dx + HALO;
            float v0 = tile[vr + 0][vc];
            float v1 = tile[vr + 1][vc];
            float v2 = tile[vr + 2][vc];
            float v3 = tile[vr + 3][vc];
            num0 = fmaf(w0, v0, num0); den0 += w0;
            num1 = fmaf(w1, v1, num1); den1 += w1;
            num2 = fmaf(w2, v2, num2); den2 += w2;
            num3 = fmaf(w3, v3, num3); den3 += w3;
            par ^= 1;
            // No barrier here: the next S12 writes the OTHER rs buffer, and
            // the barrier after it fences these S3 reads before this buffer
            // is overwritten two shifts from now.
        }
    }

    float* dst = out + (size_t)img * (IMG * IMG);
    int gx = x0 + lx;
    dst[(size_t)(y0 + ys + 0) * IMG + gx] = num0 / den0;
    dst[(size_t)(y0 + ys + 1) * IMG + gx] = num1 / den1;
    dst[(size_t)(y0 + ys + 2) * IMG + gx] = num2 / den2;
    dst[(size_t)(y0 + ys + 3) * IMG + gx] = num3 / den3;
}

extern "C" void kernel_launch(void* const* d_in, const int* in_sizes, int n_in,
                              void* d_out, int out_size, void* d_ws, size_t ws_size,
                              hipStream_t stream) {
    const float* x  = (const float*)d_in[0];
    float* out      = (float*)d_out;
    (void)in_sizes; (void)n_in; (void)out_size; (void)d_ws; (void)ws_size;
    dim3 grid(IMG / TILE_W, IMG / TILE_H, 8 * 4);
    nlm2d_kernel<<<grid, NTHREADS, 0, stream>>>(x, out);
}